// HungarianMatcher_63359357550935
// MI455X (gfx1250) — compile-verified
//
#include <hip/hip_runtime.h>
#include <stdint.h>
#include <math.h>

// Problem sizes (fixed by setup_inputs)
#define BSZ 16
#define NQ 300
#define NT_PER_B 50
#define NT (BSZ * NT_PER_B)   // 800 total targets
#define K3 51                 // 17 keypoints * 3
#define FSTRIDE 58            // packed target feature floats: 51 kp + 6 box + vol
#define TT 256                // targets staged per LDS tile
#define QB 16                 // queries per block in cost kernel
#define NROWS (BSZ * NQ)      // 4800 cost rows

typedef unsigned int u32x4 __attribute__((ext_vector_type(4)));
typedef int i32x4 __attribute__((ext_vector_type(4)));
typedef int i32x8 __attribute__((ext_vector_type(8)));

#if defined(__has_builtin) && __has_builtin(__builtin_amdgcn_tensor_load_to_lds)
#define HAVE_TDM 1
#else
#define HAVE_TDM 0
#endif

#if HAVE_TDM
// 1-D TDM load: nelem_tile contiguous f32 from gptr into LDS byte offset ldsOff.
// D# per CDNA5 ISA: group0 {count=1, lds_addr, global_addr,
// type=2}; group1 {data_size=4B, tensor_dim0=nelem_tensor, tensor_dim1=1,
// tile_dim0=nelem_tile, tensor_dim0_stride=nelem_tensor}. Tracked by TENSORcnt.
__device__ __forceinline__ void tdm_load_1d(uint32_t ldsOff, const void* gptr,
                                            uint32_t nelem_tile, uint32_t nelem_tensor)
{
    uint64_t gaddr = (uint64_t)(uintptr_t)gptr;
    u32x4 g0;
    g0.x = 1u;                                           // count=1 (valid user D#)
    g0.y = ldsOff;                                       // lds_addr (bytes)
    g0.z = (uint32_t)gaddr;                              // global_addr[31:0]
    g0.w = (uint32_t)((gaddr >> 32) & 0x01FFFFFFu)       // global_addr[56:32]
         | (2u << 30);                                   // type = 2 ("image")
    i32x8 g1;
    g1[0] = (int)(2u << 16);                             // data_size = 2 -> 4 bytes
    g1[1] = (int)((nelem_tensor & 0xFFFFu) << 16);       // tensor_dim0[15:0]
    g1[2] = (int)(((nelem_tensor >> 16) & 0xFFFFu)       // tensor_dim0[31:16]
         | (1u << 16));                                  // tensor_dim1 = 1
    g1[3] = (int)(nelem_tile << 16);                     // tile_dim0 (16-bit)
    g1[4] = 0;                                           // tile_dim1/2 unused
    g1[5] = (int)nelem_tensor;                           // tensor_dim0_stride lo32
    g1[6] = 0;
    g1[7] = 0;
    i32x4 g2 = {0, 0, 0, 0};
    i32x4 g3 = {0, 0, 0, 0};
#if __has_include(<hip/amd_detail/amd_gfx1250_TDM.h>)
    i32x8 g4 = {0, 0, 0, 0, 0, 0, 0, 0};
    __builtin_amdgcn_tensor_load_to_lds(g0, g1, g2, g3, g4, 0);
#else
    __builtin_amdgcn_tensor_load_to_lds(g0, g1, g2, g3, 0);
#endif
}
#endif

// ---------------------------------------------------------------------------
// Kernel 0: pack per-target features: dist_kp[51], xyz_min[3], xyz_max[3], vol
// ---------------------------------------------------------------------------
__global__ void pack_targets_kernel(const float* __restrict__ tgt_boxes,
                                    const float* __restrict__ tgt_kp,
                                    float* __restrict__ feat)
{
    int t = blockIdx.x * blockDim.x + threadIdx.x;
    if (t >= NT) return;
    const float* bx = tgt_boxes + t * 6;
    float cx = bx[0], cy = bx[1], cz = bx[2];
    float w = bx[3], h = bx[4], d = bx[5];
    float x0 = cx - 0.5f * w, y0 = cy - 0.5f * h, z0 = cz - 0.5f * d;
    float x1 = cx + 0.5f * w, y1 = cy + 0.5f * h, z1 = cz + 0.5f * d;
    float invw = 1.0f / w, invh = 1.0f / h, invd = 1.0f / d;
    float* f = feat + (size_t)t * FSTRIDE;
    const float* kp = tgt_kp + (size_t)t * K3;
    for (int k = 0; k < 17; ++k) {
        f[k * 3 + 0] = (kp[k * 3 + 0] - x0) * invw;
        f[k * 3 + 1] = (kp[k * 3 + 1] - y0) * invh;
        f[k * 3 + 2] = (kp[k * 3 + 2] - z0) * invd;
    }
    f[51] = x0; f[52] = y0; f[53] = z0;
    f[54] = x1; f[55] = y1; f[56] = z1;
    f[57] = w * h * d;
}

// ---------------------------------------------------------------------------
// Kernel 1: cost matrix. Block = 256 threads (8 wave32).
//   grid.x: query chunks of QB, grid.y: target tiles of TT.
//   Both the target-feature tile and the query-keypoint tile are DMA'd into
//   LDS by the Tensor Data Mover; lane <-> target column so C stores coalesce.
// ---------------------------------------------------------------------------
__global__ __launch_bounds__(256)
void cost_kernel(const float* __restrict__ pred_kp,
                 const float* __restrict__ pred_boxes,
                 const float* __restrict__ feat,
                 float* __restrict__ C)
{
    __shared__ float tf[TT * FSTRIDE];   // 59392 B target tile
    __shared__ float qkp[QB * K3];       //  3264 B query keypoints (dense)
    __shared__ float qbox[QB * 8];       //   512 B query box features

    const int tid = threadIdx.x;
    const int tileT = blockIdx.y * TT;
    const int qbase = blockIdx.x * QB;

#if HAVE_TDM
    // ---- CDNA5 TDM: two async DMA loads, one TENSORcnt wait ----
    if (tid < 32) {
        tdm_load_1d((uint32_t)(uintptr_t)(&tf[0]),
                    feat + (size_t)tileT * FSTRIDE,
                    (uint32_t)(TT * FSTRIDE),                       // 14848 < 2^16
                    (uint32_t)(NT * FSTRIDE - tileT * FSTRIDE));    // OOB tail -> 0
        tdm_load_1d((uint32_t)(uintptr_t)(&qkp[0]),
                    pred_kp + (size_t)qbase * K3,
                    (uint32_t)(QB * K3),                            // 816 contiguous
                    (uint32_t)(QB * K3));
        __builtin_amdgcn_s_wait_tensorcnt(0);
    }
#else
    for (int i = tid; i < TT * FSTRIDE; i += 256) {
        int g = tileT * FSTRIDE + i;
        tf[i] = (g < NT * FSTRIDE) ? feat[g] : 0.0f;
    }
    for (int i = tid; i < QB * K3; i += 256)
        qkp[i] = pred_kp[(size_t)qbase * K3 + i];
#endif

    // ---- transform QB query boxes (cxcyczwhd -> xyzxyz + vol) ----
    if (tid < QB) {
        const float* bx = pred_boxes + (size_t)(qbase + tid) * 6;
        float cx = bx[0], cy = bx[1], cz = bx[2];
        float w = bx[3], h = bx[4], d = bx[5];
        float* f = qbox + tid * 8;
        f[0] = cx - 0.5f * w; f[1] = cy - 0.5f * h; f[2] = cz - 0.5f * d;
        f[3] = cx + 0.5f * w; f[4] = cy + 0.5f * h; f[5] = cz + 0.5f * d;
        f[6] = w * h * d;
    }
    __syncthreads();

    const int tGlob = tileT + tid;
    const bool tOK = (tGlob < NT);
    const float* tfe = &tf[tid * FSTRIDE];
    const float t0 = tfe[51], t1 = tfe[52], t2 = tfe[53];
    const float t3 = tfe[54], t4 = tfe[55], t5 = tfe[56];
    const float vol2 = tfe[57];

    for (int qq = 0; qq < QB; ++qq) {
        const float* fq = &qkp[qq * K3];
        // L1 keypoint cost
        float ckp = 0.0f;
        #pragma unroll 3
        for (int k = 0; k < K3; ++k) ckp += fabsf(fq[k] - tfe[k]);
        // Generalized 3D IoU
        const float* fb = &qbox[qq * 8];
        float q0 = fb[0], q1 = fb[1], q2 = fb[2];
        float q3 = fb[3], q4 = fb[4], q5 = fb[5];
        float vol1 = fb[6];
        float ix = fminf(q3, t3) - fmaxf(q0, t0);
        float iy = fminf(q4, t4) - fmaxf(q1, t1);
        float iz = fminf(q5, t5) - fmaxf(q2, t2);
        float inter = fmaxf(ix, 0.0f) * fmaxf(iy, 0.0f) * fmaxf(iz, 0.0f);
        float uni = vol1 + vol2 - inter;
        float iou = inter / uni;
        float ex = fmaxf(q3, t3) - fminf(q0, t0);
        float ey = fmaxf(q4, t4) - fminf(q1, t1);
        float ez = fmaxf(q5, t5) - fminf(q2, t2);
        float evol = fmaxf(ex, 0.0f) * fmaxf(ey, 0.0f) * fmaxf(ez, 0.0f);
        float giou = iou - (evol - uni) / evol;
        if (tOK) C[(size_t)(qbase + qq) * NT + tGlob] = ckp - giou;
    }
}

// ---------------------------------------------------------------------------
// Kernel 2: per-batch Hungarian (Jonker–Volgenant, exact mirror of reference
// _lap on cost.T = [50 x 300]). One wave32 per batch; lanes parallelize the
// 300-column relax/argmin; butterfly shfl_xor reduction (wave32-safe).
// ---------------------------------------------------------------------------
__global__ __launch_bounds__(32)
void lap_kernel(const float* __restrict__ C, float* __restrict__ idxOut)
{
    const int n = NT_PER_B;   // 50 rows (targets)
    const int m = NQ;         // 300 cols (queries)
    const int b = blockIdx.x;
    const int lane = threadIdx.x;

    __shared__ float u[NT_PER_B], v[NQ], shortest[NQ];
    __shared__ int path[NQ], SCv[NQ], row4col[NQ], col4row[NT_PER_B], SRv[NT_PER_B];

    // Warm L2/WGP$ with this batch's 300x50 cost block before the serial phase
    // (global_prefetch_b8; no counters, fire-and-forget).
    for (int j = lane; j < m; j += 32)
        __builtin_prefetch(&C[(size_t)(b * NQ + j) * NT + b * NT_PER_B], 0, 3);

    for (int j = lane; j < m; j += 32) { v[j] = 0.0f; row4col[j] = -1; }
    for (int i = lane; i < n; i += 32) { u[i] = 0.0f; col4row[i] = -1; }
    __syncthreads();

    for (int curRow = 0; curRow < n; ++curRow) {
        for (int j = lane; j < m; j += 32) { shortest[j] = INFINITY; path[j] = -1; SCv[j] = 0; }
        for (int i = lane; i < n; i += 32) SRv[i] = 0;
        __syncthreads();

        float minVal = 0.0f;
        int i = curRow;
        int sink = -1;
        while (sink < 0) {
            if (lane == 0) SRv[i] = 1;
            float ui = u[i];
            for (int j = lane; j < m; j += 32) {
                if (!SCv[j]) {
                    // cost.T[i][j] = C[b*NQ + j][b*NT_PER_B + i]
                    float r = minVal + C[(size_t)(b * NQ + j) * NT + b * NT_PER_B + i] - ui - v[j];
                    if (r < shortest[j]) { shortest[j] = r; path[j] = i; }
                }
            }
            __syncthreads();
            float best = INFINITY; int bj = -1;
            for (int j = lane; j < m; j += 32)
                if (!SCv[j] && shortest[j] < best) { best = shortest[j]; bj = j; }
            for (int off = 16; off > 0; off >>= 1) {
                float ob = __shfl_xor(best, off, 32);
                int   oj = __shfl_xor(bj, off, 32);
                if (ob < best || (ob == best && (unsigned)oj < (unsigned)bj)) { best = ob; bj = oj; }
            }
            int j = bj;
            minVal = best;
            if (lane == 0) SCv[j] = 1;
            __syncthreads();
            int r4c = row4col[j];
            if (r4c == -1) sink = j; else i = r4c;
            __syncthreads();
        }
        // dual updates (before augmentation, as in reference)
        for (int i2 = lane; i2 < n; i2 += 32)
            if (SRv[i2]) u[i2] += (i2 == curRow) ? minVal : (minVal - shortest[col4row[i2]]);
        for (int j2 = lane; j2 < m; j2 += 32)
            if (SCv[j2]) v[j2] -= minVal - shortest[j2];
        __syncthreads();
        // augment along path
        if (lane == 0) {
            int j = sink;
            for (;;) {
                int pi = path[j];
                row4col[j] = pi;
                int tmp = col4row[pi];
                col4row[pi] = j;
                j = tmp;
                if (pi == curRow) break;
            }
        }
        __syncthreads();
    }

    // emit (query-sorted) index pairs as floats: indices[b][0][:]=queries, [1][:]=targets
    for (int t = lane; t < n; t += 32) {
        int q = col4row[t];
        int pos = 0;
        for (int t2 = 0; t2 < n; ++t2) pos += (col4row[t2] < q) ? 1 : 0;
        idxOut[(size_t)b * (2 * NT_PER_B) + pos] = (float)q;
        idxOut[(size_t)b * (2 * NT_PER_B) + NT_PER_B + pos] = (float)t;
    }
}

// ---------------------------------------------------------------------------
extern "C" void kernel_launch(void* const* d_in, const int* in_sizes, int n_in,
                              void* d_out, int out_size, void* d_ws, size_t ws_size,
                              hipStream_t stream)
{
    (void)in_sizes; (void)n_in; (void)out_size; (void)ws_size;
    const float* pred_kp    = (const float*)d_in[0];  // (16,300,51)
    const float* pred_boxes = (const float*)d_in[1];  // (16,300,6)
    const float* tgt_boxes  = (const float*)d_in[2];  // (16,50,6)
    const float* tgt_kp     = (const float*)d_in[3];  // (16,50,51)

    float* C = (float*)d_out;                         // 4800 x 800
    float* idxOut = C + (size_t)NROWS * NT;           // 16 x 2 x 50 (as float)
    float* feat = (float*)d_ws;                       // 800 x 58 packed target features

    pack_targets_kernel<<<(NT + 255) / 256, 256, 0, stream>>>(tgt_boxes, tgt_kp, feat);

    dim3 grid(NROWS / QB, (NT + TT - 1) / TT);
    cost_kernel<<<grid, 256, 0, stream>>>(pred_kp, pred_boxes, feat, C);

    lap_kernel<<<BSZ, 32, 0, stream>>>(C, idxOut);
}